// SingleQubitOperationLayer_27470610825416
// MI455X (gfx1250) — compile-verified
//
#include <hip/hip_runtime.h>
#include <math.h>

// ---------------------------------------------------------------------------
// SingleQubitOperationLayer on MI455X (gfx1250, wave32, WMMA f32 16x16x4)
//
// out = (((z . d_in) @ B) . d_theta @ B) . d_phi,  B = kron^12(BS), N=4096
//
// B = B6 (x) B6 with B6[r][c] = 0.125 * i^(popcount(r^c)&3): reshape z as a
// 64x64 matrix Z and each application of B is B6 @ Z @ B6 (B6 symmetric).
// B6 is generated in registers (popcount), so HBM traffic is just x (2MB) +
// out (2MB) -> memory-bound at 23.3 TB/s. Complex 64x64 matmuls run on
// V_WMMA_F32_16X16X4_F32 (f32 keeps reference complex64 precision).
//
// Round-2 changes: 512 threads = 16 waves, ONE tile per wave per stage and
// K-loop unroll 4 (keeps VGPRs < 256: no s_set_vgpr_msb, better occupancy);
// phase diagonals split into hi(row)+lo(col) parts with compile-time /
// wave-uniform selector bits folded out of the per-element path.
// ---------------------------------------------------------------------------

typedef __attribute__((ext_vector_type(2))) float v2f;
typedef __attribute__((ext_vector_type(8))) float v8f;

#define DIMN 4096
#define HALF 64

static __device__ __forceinline__ v8f wmma_f32(v2f a, v2f b, v8f c) {
  // D = A(16x4) x B(4x16) + C(16x16), all f32.
  return __builtin_amdgcn_wmma_f32_16x16x4_f32(
      /*neg_a=*/false, a, /*neg_b=*/false, b,
      /*c_mod=*/(short)0, c, /*reuse_a=*/false, /*reuse_b=*/false);
}

// B6[r][c] = 0.125 * i^(popcount(r^c) & 3)
static __device__ __forceinline__ void b6_entry(int r, int c, float& re, float& im) {
  const float S = 0.125f;  // (1/sqrt2)^6
  int p = __builtin_popcount((unsigned)(r ^ c)) & 3;
  re = (p == 0) ? S : ((p == 2) ? -S : 0.0f);
  im = (p == 1) ? S : ((p == 3) ? -S : 0.0f);
}

// qubit q term: bit ? B[q] : A[q]   (B == nullptr -> 0; folds when inlined)
static __device__ __forceinline__ float selq(int bit, const float* __restrict__ A,
                                             const float* __restrict__ B, int q) {
  float av = A[q];
  float bv = (B != nullptr) ? B[q] : 0.0f;
  return bit ? bv : av;
}

// low half of the phase: qubits 6..11, driven by bits 5..0 of c (= j & 63)
static __device__ __forceinline__ float lo_angle(int c, const float* __restrict__ A,
                                                 const float* __restrict__ B) {
  float s = 0.0f;
#pragma unroll
  for (int t = 0; t < 6; ++t) s += selq((c >> (5 - t)) & 1, A, B, 6 + t);
  return s;
}

// T = B6 @ Z   (complex).  A = B6 generated in registers, B = Z from LDS.
// One 16x16 output tile per wave (16 waves == 16 tiles).
static __device__ __forceinline__ void left_mul_b6(const float* __restrict__ Zr,
                                                   const float* __restrict__ Zi,
                                                   float* __restrict__ Tr,
                                                   float* __restrict__ Ti,
                                                   int lane, int wave) {
  const int tm = (wave >> 2) * 16;
  const int tn = (wave & 3) * 16;
  v8f cre = {};
  v8f cim = {};
  const int mA = tm + (lane & 15);  // A-frag row (ISA 16x4 f32 layout)
  const int nB = tn + (lane & 15);  // B-frag col
  const int laneK = (lane >> 4) << 1;
#pragma unroll 4
  for (int k0 = 0; k0 < HALF; k0 += 4) {
    const int kb = k0 + laneK;  // lanes 0-15: K=k0,k0+1; lanes 16-31: k0+2,k0+3
    float r0, i0, r1, i1;
    b6_entry(mA, kb, r0, i0);
    b6_entry(mA, kb + 1, r1, i1);
    v2f aP;  aP.x = r0;   aP.y = r1;   // Re(B6)
    v2f aQ;  aQ.x = i0;   aQ.y = i1;   // Im(B6)
    v2f aQn; aQn.x = -i0; aQn.y = -i1; // -Im(B6)
    v2f bX, bY;
    bX.x = Zr[kb * HALF + nB]; bX.y = Zr[(kb + 1) * HALF + nB];
    bY.x = Zi[kb * HALF + nB]; bY.y = Zi[(kb + 1) * HALF + nB];
    // Cre += P*X - Q*Y ;  Cim += P*Y + Q*X
    cre = wmma_f32(aP, bX, cre);
    cre = wmma_f32(aQn, bY, cre);
    cim = wmma_f32(aP, bY, cim);
    cim = wmma_f32(aQ, bX, cim);
  }
  // C/D layout: VGPR v -> row tm+v (lanes 0-15) / tm+v+8 (lanes 16-31)
  const int col = tn + (lane & 15);
  const int rbase = tm + ((lane >> 4) << 3);
#pragma unroll
  for (int v = 0; v < 8; ++v) {
    Tr[(rbase + v) * HALF + col] = cre[v];
    Ti[(rbase + v) * HALF + col] = cim[v];
  }
}

// Out = (T @ B6) . exp(i*phase)   (complex).  A = T from LDS, B = B6 in regs.
// Stores to LDS (outR/outI) or, if gout != nullptr, straight to global
// (gout[j] = re, gout[DIMN + j] = im).
static __device__ __forceinline__ void right_mul_b6(const float* __restrict__ Ar,
                                                    const float* __restrict__ Ai,
                                                    float* __restrict__ outR,
                                                    float* __restrict__ outI,
                                                    float* __restrict__ gout,
                                                    const float* __restrict__ angA,
                                                    const float* __restrict__ angB,
                                                    int lane, int wave) {
  const int tm = (wave >> 2) * 16;
  const int tn = (wave & 3) * 16;
  v8f cre = {};
  v8f cim = {};
  const int mA = tm + (lane & 15);
  const int nB = tn + (lane & 15);
  const int laneK = (lane >> 4) << 1;
#pragma unroll 4
  for (int k0 = 0; k0 < HALF; k0 += 4) {
    const int kb = k0 + laneK;
    v2f aX, aY;
    aX.x = Ar[mA * HALF + kb]; aX.y = Ar[mA * HALF + kb + 1];
    aY.x = Ai[mA * HALF + kb]; aY.y = Ai[mA * HALF + kb + 1];
    float r0, i0, r1, i1;
    b6_entry(kb, nB, r0, i0);
    b6_entry(kb + 1, nB, r1, i1);
    v2f bP;  bP.x = r0;   bP.y = r1;
    v2f bQ;  bQ.x = i0;   bQ.y = i1;
    v2f bQn; bQn.x = -i0; bQn.y = -i1;
    // Cre += X*P - Y*Q ;  Cim += X*Q + Y*P
    cre = wmma_f32(aX, bP, cre);
    cre = wmma_f32(aY, bQn, cre);
    cim = wmma_f32(aX, bQ, cim);
    cim = wmma_f32(aY, bP, cim);
  }
  const int col = tn + (lane & 15);
  const int rbase = tm + ((lane >> 4) << 3);
  // Phase epilogue: angle(row,col) = hi(row) + lo(col).
  // row = rbase + v: bits 5,4 of row come from (wave>>2) [wave-uniform],
  // bit 3 from (lane>>4), bits 2..0 are the compile-time unroll index v.
  const int trow = wave >> 2;
  float angBase = selq((trow >> 1) & 1, angA, angB, 0)
                + selq(trow & 1, angA, angB, 1)
                + selq((lane >> 4) & 1, angA, angB, 2)
                + lo_angle(col, angA, angB);
#pragma unroll
  for (int v = 0; v < 8; ++v) {
    const int j = (rbase + v) * HALF + col;
    // v bits are compile-time: selq folds to a plain scalar add.
    const float ang = angBase + selq((v >> 2) & 1, angA, angB, 3)
                              + selq((v >> 1) & 1, angA, angB, 4)
                              + selq(v & 1, angA, angB, 5);
    float sn, cs;
    __sincosf(ang, &sn, &cs);
    const float re2 = cre[v] * cs - cim[v] * sn;
    const float im2 = cre[v] * sn + cim[v] * cs;
    if (gout != nullptr) {
      gout[j] = re2;
      gout[DIMN + j] = im2;
    } else {
      outR[j] = re2;
      outI[j] = im2;
    }
  }
}

__global__ __launch_bounds__(512) void sq_layer_kernel(
    const float* __restrict__ x, const float* __restrict__ alphas,
    const float* __restrict__ betas, const float* __restrict__ thetas,
    const float* __restrict__ phis, float* __restrict__ out) {
  __shared__ float Zre[HALF * HALF];
  __shared__ float Zim[HALF * HALF];
  __shared__ float Tre[HALF * HALF];
  __shared__ float Tim[HALF * HALF];  // 4 * 16KB = 64KB LDS

  const int b = blockIdx.x;
  const int tid = (int)threadIdx.x;
  const int lane = tid & 31;  // wave32
  const int wave = tid >> 5;  // 16 waves == 16 output tiles

  const float* __restrict__ x0 = x + (size_t)b * 2 * DIMN;
  const float* __restrict__ x1 = x0 + DIMN;

  // Load z = x0 + i*x1 with fused d_in phase. j = e*512 + tid:
  //   c = j & 63 = tid & 63 (constant across e)  -> lo part once per thread
  //   r = j >> 6 = e*8 + (tid>>6): bits 5..3 of r = e (compile-time),
  //   bits 2..0 = tid>>6 (fixed per thread).
  const int rlo = tid >> 6;  // 0..7
  const float angThread = selq((rlo >> 2) & 1, alphas, betas, 3)
                        + selq((rlo >> 1) & 1, alphas, betas, 4)
                        + selq(rlo & 1, alphas, betas, 5)
                        + lo_angle(tid & 63, alphas, betas);
#pragma unroll
  for (int e = 0; e < DIMN / 512; ++e) {
    const int j = e * 512 + tid;
    const float xr = x0[j];
    const float xi = x1[j];
    // e bits are compile-time: selq folds to scalar adds.
    const float ang = angThread + selq((e >> 2) & 1, alphas, betas, 0)
                                + selq((e >> 1) & 1, alphas, betas, 1)
                                + selq(e & 1, alphas, betas, 2);
    float sn, cs;
    __sincosf(ang, &sn, &cs);
    Zre[j] = xr * cs - xi * sn;
    Zim[j] = xr * sn + xi * cs;
  }
  __syncthreads();

  // Application 1 of B:  Z <- B6 @ Z @ B6, fuse d_theta into the epilogue.
  left_mul_b6(Zre, Zim, Tre, Tim, lane, wave);
  __syncthreads();
  right_mul_b6(Tre, Tim, Zre, Zim, /*gout=*/nullptr, thetas, /*angB=*/nullptr, lane, wave);
  __syncthreads();

  // Application 2 of B, fuse d_phi + write directly to global output.
  left_mul_b6(Zre, Zim, Tre, Tim, lane, wave);
  __syncthreads();
  right_mul_b6(Tre, Tim, /*outR=*/nullptr, /*outI=*/nullptr,
               out + (size_t)b * 2 * DIMN, phis, /*angB=*/nullptr, lane, wave);
}

extern "C" void kernel_launch(void* const* d_in, const int* in_sizes, int n_in,
                              void* d_out, int out_size, void* d_ws, size_t ws_size,
                              hipStream_t stream) {
  const float* x      = (const float*)d_in[0];  // (64, 2, 4096) f32
  const float* alphas = (const float*)d_in[1];  // (12,) f32
  const float* betas  = (const float*)d_in[2];
  const float* thetas = (const float*)d_in[3];
  const float* phis   = (const float*)d_in[4];
  float* out = (float*)d_out;                   // (64, 2, 4096) f32

  const int batch = in_sizes[0] / (2 * DIMN);   // 64
  sq_layer_kernel<<<dim3(batch), dim3(512), 0, stream>>>(x, alphas, betas, thetas, phis, out);
}